// LIFSpike_52012053954601
// MI455X (gfx1250) — compile-verified
//
#include <hip/hip_runtime.h>

// LIF (leaky integrate-and-fire) scan over T=16 timesteps.
// x, out: [B=32, T=16, C=128, H=32, W=32] float32.
// Per spatial site: mem = mem*TAU + x_t ; s = (mem >= THRESH) ; mem = s ? 0 : mem.
// Pure streaming op: 268 MB in + 268 MB out, zero reuse -> HBM-bound (~23 us @ 23.3 TB/s).

typedef float v4f __attribute__((ext_vector_type(4)));

constexpr int   T_STEPS = 16;
constexpr int   CHW     = 128 * 32 * 32;   // 131072 floats per (b, t) slab
constexpr int   CHW4    = CHW / 4;         // 32768 float4 per (b, t) slab
constexpr float TAU     = 0.25f;
constexpr float THRESH  = 0.5f;

__global__ __launch_bounds__(256) void lif_scan_kernel(const v4f* __restrict__ x,
                                                       v4f* __restrict__ out,
                                                       int nvec_total /* B * CHW4 */)
{
    const int idx = blockIdx.x * blockDim.x + threadIdx.x;
    if (idx >= nvec_total) return;

    // idx -> (batch, vec-within-slab); CHW4 is a power of two -> shifts.
    const int  b    = idx / CHW4;
    const int  v    = idx - b * CHW4;
    const long base = (long)b * ((long)T_STEPS * CHW4) + v;   // float4 units

    // Stage all 16 timestep loads up front: independent addresses, 16 b128
    // loads in flight per thread (non-temporal: stream > L2, no reuse).
    v4f xs[T_STEPS];
#pragma unroll
    for (int t = 0; t < T_STEPS; ++t) {
        xs[t] = __builtin_nontemporal_load(x + base + (long)t * CHW4);
    }

    // Serial scan in registers; 4 sites per thread.
    float mem[4] = {0.0f, 0.0f, 0.0f, 0.0f};

#pragma unroll
    for (int t = 0; t < T_STEPS; ++t) {
        v4f o;
#pragma unroll
        for (int j = 0; j < 4; ++j) {
            float m   = __builtin_fmaf(mem[j], TAU, xs[t][j]);  // leaky integrate
            bool  fire = (m >= THRESH);                          // mem - THRESH >= 0
            o[j]      = fire ? 1.0f : 0.0f;                      // spike out
            mem[j]    = fire ? 0.0f : m;                         // hard reset
        }
        __builtin_nontemporal_store(o, out + base + (long)t * CHW4);
    }
}

extern "C" void kernel_launch(void* const* d_in, const int* in_sizes, int n_in,
                              void* d_out, int out_size, void* d_ws, size_t ws_size,
                              hipStream_t stream)
{
    const float* x   = (const float*)d_in[0];
    float*       out = (float*)d_out;

    // Total float4 "sites" across batch = total_elems / (T * 4).
    const int total_elems = in_sizes[0];                 // 67,108,864
    const int nvec_total  = total_elems / (T_STEPS * 4); // 1,048,576

    const int block = 256;                               // 8 wave32 per workgroup
    const int grid  = (nvec_total + block - 1) / block;  // 4096 blocks

    lif_scan_kernel<<<grid, block, 0, stream>>>((const v4f*)x, (v4f*)out, nvec_total);
}